// ODEVecField_70626442216096
// MI455X (gfx1250) — compile-verified
//
#include <hip/hip_runtime.h>

// Grid / moment dimensions (fixed by the reference)
#define NYg 64
#define NXg 64
#define NZg 32
#define G   131072          // NYg*NXg*NZg = 2^17
#define NS  2
#define NP  4
#define NM  4
#define NN  4
#define NMOM 128            // NS*NP*NM*NN
#define NFLD 6
#define PI_F 3.14159265358979323846f

typedef __attribute__((ext_vector_type(2))) float v2f;
typedef __attribute__((ext_vector_type(8))) float v8f;
typedef __attribute__((ext_vector_type(4))) unsigned int u32x4;
typedef __attribute__((ext_vector_type(4))) int i32x4;
typedef __attribute__((ext_vector_type(8))) int i32x8;

struct c2 { float re, im; };
__device__ __forceinline__ c2 cadd(c2 a, c2 b){ return { a.re+b.re, a.im+b.im }; }
__device__ __forceinline__ c2 csub(c2 a, c2 b){ return { a.re-b.re, a.im-b.im }; }
__device__ __forceinline__ c2 cmul(c2 a, c2 b){ return { a.re*b.re - a.im*b.im, a.re*b.im + a.im*b.re }; }
__device__ __forceinline__ c2 cscale(float s, c2 a){ return { s*a.re, s*a.im }; }

__device__ __forceinline__ v8f cwmma(v2f a, v2f b, v8f c) {
  // D = A(16x4,f32) * B(4x16,f32) + C(16x16,f32)  -> v_wmma_f32_16x16x4_f32
  return __builtin_amdgcn_wmma_f32_16x16x4_f32(false, a, false, b, (short)0, c,
                                               false, false);
}

#if defined(__HIP_DEVICE_COMPILE__) && __has_builtin(__builtin_amdgcn_tensor_load_to_lds) && __has_builtin(__builtin_amdgcn_s_wait_tensorcnt)
#define HAVE_TDM 1
#else
#define HAVE_TDM 0
#endif

// ---------------------------------------------------------------------------
// Stage the twiddle matrix (re plane followed contiguously by im plane,
// 2*w_count dwords total) into LDS. One wave per block issues a Tensor Data
// Mover descriptor (1-row tile); fallback is a cooperative vector copy.
// ---------------------------------------------------------------------------
__device__ __forceinline__ void stage_w_lds(float* sW,
                                            const float* __restrict__ wre,
                                            const float* __restrict__ wim,
                                            int w_count)
{
#if HAVE_TDM
  if (threadIdx.x < 32) {                      // one wave drives the TDM
    const unsigned lds_base = (unsigned)(uintptr_t)(void*)sW;   // low 32 bits = LDS offset
    const unsigned long long ga = (unsigned long long)(uintptr_t)wre;
    const unsigned ne = 2u * (unsigned)w_count;                 // dwords to move

    u32x4 g0;
    g0.x = 1u;                                  // count=1 (valid user descriptor)
    g0.y = lds_base;                            // lds_addr (bytes)
    g0.z = (unsigned)(ga & 0xFFFFFFFFull);      // global_addr[31:0]
    g0.w = (unsigned)((ga >> 32) & 0x01FFFFFFull) | (2u << 30); // addr[56:32] | type=2

    i32x8 g1;
    g1[0] = (int)(2u << 16);                    // workgroup_mask=0, data_size=4B
    g1[1] = (int)((ne & 0xFFFFu) << 16);        // tensor_dim0[15:0]
    g1[2] = (int)(((ne >> 16) & 0xFFFFu) | (1u << 16)); // tensor_dim0[31:16] | tensor_dim1=1
    g1[3] = (int)((ne & 0xFFFFu) << 16);        // tensor_dim1[31:16]=0 | tile_dim0
    g1[4] = 1;                                  // tile_dim1=1, tile_dim2=0
    g1[5] = (int)ne;                            // tensor_dim0_stride[31:0]
    g1[6] = 0;                                  // stride hi / dim1_stride lo
    g1[7] = 0;

    i32x4 z4 = { 0, 0, 0, 0 };
#if defined(__clang_major__) && (__clang_major__ >= 23)
    i32x8 z8 = {};
    __builtin_amdgcn_tensor_load_to_lds(g0, g1, z4, z4, z8, 0);
#else
    __builtin_amdgcn_tensor_load_to_lds(g0, g1, z4, z4, 0);
#endif
    __builtin_amdgcn_s_wait_tensorcnt(0);
  }
  __syncthreads();
#else
  for (int i = threadIdx.x; i < w_count; i += 256) {
    sW[i]           = wre[i];
    sW[w_count + i] = wim[i];
  }
  __syncthreads();
#endif
}

// ---------------------------------------------------------------------------
// Generic strided batched complex GEMM:  O[b,m,n] = sum_k A[b,m,k]*B[b,k,n]
// (complex split re/im planes; strides in complex elements)
// WA=1: A is the (LDS-staged) twiddle matrix; WA=0: B is.
// AK1=1: A data is contiguous in k (vector b64 loads).
// Each wave computes MREP x NREP 16x16 complex tiles, reusing the streamed
// operand across the replicated direction.
// ---------------------------------------------------------------------------
template<int MREP, int NREP, int WA, int AK1>
__global__ void __launch_bounds__(256)
cgemm_wmma(const float* __restrict__ Are, const float* __restrict__ Aim,
           const float* __restrict__ Bre, const float* __restrict__ Bim,
           float* __restrict__ Ore, float* __restrict__ Oim,
           int M, int N, int K,
           long a_batch, long a_m, long a_k,
           long b_batch, long b_k, long b_n,
           long o_batch, long o_m, long o_n,
           int w_count, int total_tiles)
{
  __shared__ float sW[8192];                    // 32 KB: re plane, then im plane
  stage_w_lds(sW, WA ? Are : Bre, WA ? Aim : Bim, w_count);
  const float* sWre = sW;
  const float* sWim = sW + w_count;

  const int wave = (int)((blockIdx.x * (long)blockDim.x + threadIdx.x) >> 5);
  const int lane = threadIdx.x & 31;
  if (wave >= total_tiles) return;             // uniform per-wave branch (EXEC stays all-1)

  const int nt = N / (16 * NREP);
  const int mt = M / (16 * MREP);
  const int b  = wave / (mt * nt);
  const int r  = wave % (mt * nt);
  const int m0 = (r / nt) * (16 * MREP);
  const int n0 = (r % nt) * (16 * NREP);

  const int half = lane >> 4;                  // K-pair select (0: K0/K1, 1: K2/K3)
  const int l    = lane & 15;                  // row (A) / col (B,C,D) within tile

  v8f cre[MREP][NREP];
  v8f cim[MREP][NREP];
  #pragma unroll
  for (int i = 0; i < MREP; ++i)
    #pragma unroll
    for (int j = 0; j < NREP; ++j) { v8f z = {}; cre[i][j] = z; cim[i][j] = z; }

  for (int k0 = 0; k0 < K; k0 += 4) {
    const int kk = k0 + 2 * half;

    v2f ar[MREP], ai[MREP], nai[MREP];
    #pragma unroll
    for (int i = 0; i < MREP; ++i) {
      const int row = m0 + i * 16 + l;
      if (WA) {                                // A = twiddle matrix from LDS
        const int widx = (int)a_m * row + (int)a_k * kk;
        ar[i].x = sWre[widx]; ar[i].y = sWre[widx + (int)a_k];
        ai[i].x = sWim[widx]; ai[i].y = sWim[widx + (int)a_k];
      } else if (AK1) {                        // A = streamed data, contiguous in k
        const long ak = a_batch * b + a_m * (long)row + (long)kk;
        ar[i] = *reinterpret_cast<const v2f*>(Are + ak);
        ai[i] = *reinterpret_cast<const v2f*>(Aim + ak);
        if (k0 + 4 < K) __builtin_prefetch(&Are[ak + 4], 0, 1);
      } else {                                 // A = streamed data, strided in k
        const long ak = a_batch * b + a_m * (long)row + a_k * (long)kk;
        ar[i].x = Are[ak]; ar[i].y = Are[ak + a_k];
        ai[i].x = Aim[ak]; ai[i].y = Aim[ak + a_k];
      }
      nai[i] = -ar[i] * 0.0f - ai[i];          // sign flip of Ai (f32 WMMA NEG is C-only)
    }

    v2f br[NREP], bi[NREP];
    #pragma unroll
    for (int j = 0; j < NREP; ++j) {
      const int col = n0 + j * 16 + l;
      if (!WA) {                               // B = twiddle matrix from LDS
        const int widx = (int)b_k * kk + (int)b_n * col;
        br[j].x = sWre[widx]; br[j].y = sWre[widx + (int)b_k];
        bi[j].x = sWim[widx]; bi[j].y = sWim[widx + (int)b_k];
      } else {                                 // B = streamed data (strided k)
        const long bk = b_batch * b + b_n * (long)col + b_k * (long)kk;
        br[j].x = Bre[bk]; br[j].y = Bre[bk + b_k];
        bi[j].x = Bim[bk]; bi[j].y = Bim[bk + b_k];
        if (k0 + 4 < K) __builtin_prefetch(&Bre[bk + 4 * b_k], 0, 1);
      }
    }

    #pragma unroll
    for (int i = 0; i < MREP; ++i)
      #pragma unroll
      for (int j = 0; j < NREP; ++j) {
        cre[i][j] = cwmma(ar[i],  br[j], cre[i][j]);
        cre[i][j] = cwmma(nai[i], bi[j], cre[i][j]);
        cim[i][j] = cwmma(ar[i],  bi[j], cim[i][j]);
        cim[i][j] = cwmma(ai[i],  br[j], cim[i][j]);
      }
  }

  // C/D layout: VGPR v holds rows m0+v (lanes 0-15) and m0+8+v (lanes 16-31), col n0+l
  #pragma unroll
  for (int i = 0; i < MREP; ++i) {
    const int mrow = m0 + i * 16 + (half ? 8 : 0);
    #pragma unroll
    for (int j = 0; j < NREP; ++j) {
      const long obase = o_batch * b + o_n * (long)(n0 + j * 16 + l);
      #pragma unroll
      for (int v = 0; v < 8; ++v) {
        const long off = obase + o_m * (long)(mrow + v);
        Ore[off] = cre[i][j][v];
        Oim[off] = cim[i][j][v];
      }
    }
  }
}

// ---------------------------------------------------------------------------
// Build shift-folded DFT matrices.
//  inverse: w[j,k] = (1/N) exp(+2*pi*i * j * (k - N/2) / N)   (ifft(ifftshift(.)))
//  forward: w[j,k] =       exp(-2*pi*i * (j + N/2) * k / N)   (fftshift(fft(.)))
// ---------------------------------------------------------------------------
__global__ void build_dft(float* __restrict__ wre, float* __restrict__ wim,
                          int N, int forward, float scale)
{
  int t = blockIdx.x * blockDim.x + threadIdx.x;
  if (t >= N * N) return;
  int j = t / N, k = t % N;
  float ph = forward
           ? (-2.0f * PI_F * (float)(j + N / 2) * (float)k / (float)N)
           : ( 2.0f * PI_F * (float)j * (float)(k - N / 2) / (float)N);
  wre[t] = scale * cosf(ph);
  wim[t] = scale * sinf(ph);
}

// ---------------------------------------------------------------------------
// Real-space nonlinear term.
// ---------------------------------------------------------------------------
__device__ __forceinline__ c2 getC(const float* __restrict__ Cre,
                                   const float* __restrict__ Cim,
                                   int s, int p, int m, int n, int g)
{
  if ((unsigned)p > 3u || (unsigned)m > 3u || (unsigned)n > 3u) return {0.f, 0.f};
  long idx = (((long)((((s * NP) + p) * NM + m) * NN + n)) << 17) + g;
  return { Cre[idx], Cim[idx] };
}

__global__ void __launch_bounds__(256)
nl_kernel(const float* __restrict__ Cre, const float* __restrict__ Cim,
          const float* __restrict__ Fre, const float* __restrict__ Fim,
          float* __restrict__ Ore, float* __restrict__ Oim,
          const float* __restrict__ qs, const float* __restrict__ alpha_s,
          const float* __restrict__ u_s,
          const float* __restrict__ snp, const float* __restrict__ snm,
          const float* __restrict__ smp, const float* __restrict__ smm,
          const float* __restrict__ spp, const float* __restrict__ spm)
{
  long tid = blockIdx.x * (long)blockDim.x + threadIdx.x;
  if (tid >= (long)NMOM * G) return;
  const int g  = (int)(tid & (G - 1));
  const int mo = (int)(tid >> 17);
  const int n = mo & 3, m = (mo >> 2) & 3, p = (mo >> 4) & 3, s = mo >> 6;

  const c2 Ex = { Fre[0 * (long)G + g], Fim[0 * (long)G + g] };
  const c2 Ey = { Fre[1 * (long)G + g], Fim[1 * (long)G + g] };
  const c2 Ez = { Fre[2 * (long)G + g], Fim[2 * (long)G + g] };
  const c2 Bx = { Fre[3 * (long)G + g], Fim[3 * (long)G + g] };
  const c2 By = { Fre[4 * (long)G + g], Fim[4 * (long)G + g] };
  const c2 Bz = { Fre[5 * (long)G + g], Fim[5 * (long)G + g] };

  const float aX = alpha_s[s * 3 + 0], aY = alpha_s[s * 3 + 1], aZ = alpha_s[s * 3 + 2];
  const float uX = u_s[s * 3 + 0],     uY = u_s[s * 3 + 1],     uZ = u_s[s * 3 + 2];
  const float q  = qs[s];

  const c2 Gx = cadd(Ex, csub(cscale(uY, Bz), cscale(uZ, By)));
  const c2 Gy = cadd(Ey, csub(cscale(uZ, Bx), cscale(uX, Bz)));
  const c2 Gz = cadd(Ez, csub(cscale(uX, By), cscale(uY, Bx)));

  // term 1: shift n -> n-1
  c2 Cn1   = getC(Cre, Cim, s, p, m, n - 1, g);
  c2 ximC1 = cadd(cscale(smp[m], getC(Cre, Cim, s, p, m + 1, n - 1, g)),
                  cscale(smm[m], getC(Cre, Cim, s, p, m - 1, n - 1, g)));
  c2 xipC1 = cadd(cscale(spp[p], getC(Cre, Cim, s, p + 1, m, n - 1, g)),
                  cscale(spm[p], getC(Cre, Cim, s, p - 1, m, n - 1, g)));
  c2 t1 = csub(cadd(cmul(Gx, Cn1), cmul(cscale(aY, Bz), ximC1)),
               cmul(cscale(aZ, By), xipC1));
  t1 = cscale(2.0f * snm[n] / aX, t1);

  // term 2: shift m -> m-1
  c2 Cm1   = getC(Cre, Cim, s, p, m - 1, n, g);
  c2 xipC2 = cadd(cscale(spp[p], getC(Cre, Cim, s, p + 1, m - 1, n, g)),
                  cscale(spm[p], getC(Cre, Cim, s, p - 1, m - 1, n, g)));
  c2 xinC2 = cadd(cscale(snp[n], getC(Cre, Cim, s, p, m - 1, n + 1, g)),
                  cscale(snm[n], getC(Cre, Cim, s, p, m - 1, n - 1, g)));
  c2 t2 = csub(cadd(cmul(Gy, Cm1), cmul(cscale(aZ, Bx), xipC2)),
               cmul(cscale(aX, Bz), xinC2));
  t2 = cscale(2.0f * smm[m] / aY, t2);

  // term 3: shift p -> p-1
  c2 Cp1   = getC(Cre, Cim, s, p - 1, m, n, g);
  c2 xinC3 = cadd(cscale(snp[n], getC(Cre, Cim, s, p - 1, m, n + 1, g)),
                  cscale(snm[n], getC(Cre, Cim, s, p - 1, m, n - 1, g)));
  c2 ximC3 = cadd(cscale(smp[m], getC(Cre, Cim, s, p - 1, m + 1, n, g)),
                  cscale(smm[m], getC(Cre, Cim, s, p - 1, m - 1, n, g)));
  c2 t3 = csub(cadd(cmul(Gz, Cp1), cmul(cscale(aX, By), xinC3)),
               cmul(cscale(aY, Bx), ximC3));
  t3 = cscale(2.0f * spm[p] / aZ, t3);

  c2 nl = cscale(q, cadd(cadd(t1, t2), t3));
  long idx = ((long)mo << 17) + g;
  Ore[idx] = nl.re;
  Oim[idx] = nl.im;
}

// ---------------------------------------------------------------------------
// dCk assembly (k-space): linear streaming + masked NL + damping.
// Output interleaved complex at d_out[0 .. 2*NMOM*G)
// ---------------------------------------------------------------------------
__global__ void __launch_bounds__(256)
dck_kernel(const float* __restrict__ Ckre, const float* __restrict__ Ckim,
           const float* __restrict__ NLre, const float* __restrict__ NLim,
           const float* __restrict__ kxg, const float* __restrict__ kyg,
           const float* __restrict__ kzg, const float* __restrict__ k2g,
           const float* __restrict__ col, const float* __restrict__ mask,
           const float* __restrict__ nu, const float* __restrict__ Dd,
           const float* __restrict__ alpha_s, const float* __restrict__ u_s,
           const float* __restrict__ snp, const float* __restrict__ snm,
           const float* __restrict__ smp, const float* __restrict__ smm,
           const float* __restrict__ spp, const float* __restrict__ spm,
           float* __restrict__ out)
{
  long tid = blockIdx.x * (long)blockDim.x + threadIdx.x;
  if (tid >= (long)NMOM * G) return;
  const int g  = (int)(tid & (G - 1));
  const int mo = (int)(tid >> 17);
  const int n = mo & 3, m = (mo >> 2) & 3, p = (mo >> 4) & 3, s = mo >> 6;

  const float aX = alpha_s[s * 3 + 0], aY = alpha_s[s * 3 + 1], aZ = alpha_s[s * 3 + 2];
  const float uX = u_s[s * 3 + 0],     uY = u_s[s * 3 + 1],     uZ = u_s[s * 3 + 2];

  const c2 Ck = getC(Ckre, Ckim, s, p, m, n, g);

  c2 xin = cadd(cscale(snp[n], getC(Ckre, Ckim, s, p, m, n + 1, g)),
                cscale(snm[n], getC(Ckre, Ckim, s, p, m, n - 1, g)));
  c2 xim = cadd(cscale(smp[m], getC(Ckre, Ckim, s, p, m + 1, n, g)),
                cscale(smm[m], getC(Ckre, Ckim, s, p, m - 1, n, g)));
  c2 xip = cadd(cscale(spp[p], getC(Ckre, Ckim, s, p + 1, m, n, g)),
                cscale(spm[p], getC(Ckre, Ckim, s, p - 1, m, n, g)));

  c2 tx = cadd(cscale(aX, xin), cscale(uX, Ck));
  c2 ty = cadd(cscale(aY, xim), cscale(uY, Ck));
  c2 tz = cadd(cscale(aZ, xip), cscale(uZ, Ck));

  const float kx = kxg[g], ky = kyg[g], kz = kzg[g];

  // -i*(kx*tx + ky*ty + kz*tz)
  c2 acc;
  acc.re =  (kx * tx.im + ky * ty.im + kz * tz.im);
  acc.im = -(kx * tx.re + ky * ty.re + kz * tz.re);

  const long idx = ((long)mo << 17) + g;
  const float mk = mask[g];
  acc.re += mk * NLre[idx];
  acc.im += mk * NLim[idx];

  const float damp = nu[s] * col[(p * NM + m) * NN + n] + Dd[s] * k2g[g];
  acc.re -= damp * Ck.re;
  acc.im -= damp * Ck.im;

  out[2 * idx + 0] = acc.re;
  out[2 * idx + 1] = acc.im;
}

// ---------------------------------------------------------------------------
// dFk: curl terms + moment currents. Output after dCk block, interleaved.
// ---------------------------------------------------------------------------
__global__ void __launch_bounds__(256)
dfk_kernel(const float* __restrict__ Fkre, const float* __restrict__ Fkim,
           const float* __restrict__ Ckre, const float* __restrict__ Ckim,
           const float* __restrict__ kxg, const float* __restrict__ kyg,
           const float* __restrict__ kzg,
           const float* __restrict__ qs, const float* __restrict__ alpha_s,
           const float* __restrict__ u_s, const float* __restrict__ Omega,
           float* __restrict__ out)
{
  int g = blockIdx.x * blockDim.x + threadIdx.x;
  if (g >= G) return;
  const float kx = kxg[g], ky = kyg[g], kz = kzg[g];

  c2 E[3], B[3];
  #pragma unroll
  for (int c = 0; c < 3; ++c) {
    E[c] = { Fkre[(long)c * G + g],       Fkim[(long)c * G + g] };
    B[c] = { Fkre[(long)(c + 3) * G + g], Fkim[(long)(c + 3) * G + g] };
  }

  c2 cE[3] = { { ky * E[2].re - kz * E[1].re, ky * E[2].im - kz * E[1].im },
               { kz * E[0].re - kx * E[2].re, kz * E[0].im - kx * E[2].im },
               { kx * E[1].re - ky * E[0].re, kx * E[1].im - ky * E[0].im } };
  c2 cB[3] = { { ky * B[2].re - kz * B[1].re, ky * B[2].im - kz * B[1].im },
               { kz * B[0].re - kx * B[2].re, kz * B[0].im - kx * B[2].im },
               { kx * B[1].re - ky * B[0].re, kx * B[1].im - ky * B[0].im } };

  const float inv_s2 = 0.70710678118654752440f;
  c2 j[3] = { {0.f,0.f}, {0.f,0.f}, {0.f,0.f} };
  #pragma unroll
  for (int s = 0; s < NS; ++s) {
    const long base = ((long)(s * 64)) << 17;            // (s,0,0,0)
    c2 C000 = { Ckre[base + g],               Ckim[base + g] };
    c2 Cx1  = { Ckre[base + ((long)1  << 17) + g], Ckim[base + ((long)1  << 17) + g] };  // n=1
    c2 Cy1  = { Ckre[base + ((long)4  << 17) + g], Ckim[base + ((long)4  << 17) + g] };  // m=1
    c2 Cz1  = { Ckre[base + ((long)16 << 17) + g], Ckim[base + ((long)16 << 17) + g] };  // p=1
    const float q = qs[s];
    j[0] = cadd(j[0], cscale(q, cadd(cscale(u_s[s*3+0], C000), cscale(alpha_s[s*3+0] * inv_s2, Cx1))));
    j[1] = cadd(j[1], cscale(q, cadd(cscale(u_s[s*3+1], C000), cscale(alpha_s[s*3+1] * inv_s2, Cy1))));
    j[2] = cadd(j[2], cscale(q, cadd(cscale(u_s[s*3+2], C000), cscale(alpha_s[s*3+2] * inv_s2, Cz1))));
  }

  const float iOm = 1.0f / Omega[0];
  const long obase = 2L * (long)NMOM * G;
  #pragma unroll
  for (int c = 0; c < 3; ++c) {
    c2 dE = { -cB[c].im - iOm * j[c].re, cB[c].re - iOm * j[c].im };   // i*cross(k,B) - j/Om
    c2 dB = { cE[c].im, -cE[c].re };                                   // -i*cross(k,E)
    long eoff = obase + 2L * ((long)c * G + g);
    long boff = obase + 2L * ((long)(c + 3) * G + g);
    out[eoff + 0] = dE.re;  out[eoff + 1] = dE.im;
    out[boff + 0] = dB.re;  out[boff + 1] = dB.im;
  }
}

// ---------------------------------------------------------------------------
// Host launcher
// ---------------------------------------------------------------------------
template<int MREP, int NREP, int WA, int AK1>
static void launch_cgemm(const float* Are, const float* Aim,
                         const float* Bre, const float* Bim,
                         float* Ore, float* Oim,
                         int M, int N, int K,
                         long ab, long am, long ak,
                         long bb, long bk, long bn,
                         long ob, long om, long on,
                         int w_count, int nbatch, hipStream_t stream)
{
  int tiles  = nbatch * (M / (16 * MREP)) * (N / (16 * NREP));
  int blocks = (tiles + 7) / 8;                 // 8 waves (tiles) per 256-thread block
  cgemm_wmma<MREP, NREP, WA, AK1><<<blocks, 256, 0, stream>>>(
      Are, Aim, Bre, Bim, Ore, Oim, M, N, K,
      ab, am, ak, bb, bk, bn, ob, om, on, w_count, tiles);
}

extern "C" void kernel_launch(void* const* d_in, const int* in_sizes, int n_in,
                              void* d_out, int out_size, void* d_ws, size_t ws_size,
                              hipStream_t stream)
{
  (void)in_sizes; (void)n_in; (void)out_size; (void)ws_size;

  const float* Ck_re = (const float*)d_in[1];
  const float* Ck_im = (const float*)d_in[2];
  const float* Fk_re = (const float*)d_in[3];
  const float* Fk_im = (const float*)d_in[4];
  const float* kxg   = (const float*)d_in[5];
  const float* kyg   = (const float*)d_in[6];
  const float* kzg   = (const float*)d_in[7];
  const float* k2g   = (const float*)d_in[8];
  const float* col   = (const float*)d_in[10];
  const float* snp   = (const float*)d_in[11];
  const float* snm   = (const float*)d_in[12];
  const float* smp   = (const float*)d_in[13];
  const float* smm   = (const float*)d_in[14];
  const float* spp   = (const float*)d_in[15];
  const float* spm   = (const float*)d_in[16];
  const float* mask  = (const float*)d_in[17];
  const float* qs    = (const float*)d_in[18];
  const float* nu    = (const float*)d_in[19];
  const float* Dd    = (const float*)d_in[20];
  const float* al    = (const float*)d_in[21];
  const float* us    = (const float*)d_in[22];
  const float* Om    = (const float*)d_in[23];
  float* out = (float*)d_out;

  // ---- workspace layout (floats); W planes contiguous re-then-im ----
  float* ws = (float*)d_ws;
  const size_t NCg = (size_t)NMOM * G;
  const size_t NFg = (size_t)NFLD * G;
  float* Are = ws;              float* Aim = Are + NCg;
  float* Bre = Aim + NCg;       float* Bim = Bre + NCg;
  float* Far = Bim + NCg;       float* Fai = Far + NFg;
  float* Fbr = Fai + NFg;       float* Fbi = Fbr + NFg;
  float* W   = Fbi + NFg;
  float* Wzi_r = W;             float* Wzi_i = Wzi_r + 1024;
  float* Wyi_r = Wzi_i + 1024;  float* Wyi_i = Wyi_r + 4096;
  float* Wxi_r = Wyi_i + 4096;  float* Wxi_i = Wxi_r + 4096;
  float* Wzf_r = Wxi_i + 4096;  float* Wzf_i = Wzf_r + 1024;
  float* Wyf_r = Wzf_i + 1024;  float* Wyf_i = Wyf_r + 4096;
  float* Wxf_r = Wyf_i + 4096;  float* Wxf_i = Wxf_r + 4096;

  // ---- build DFT matrices ----
  build_dft<<<(32*32 + 255)/256, 256, 0, stream>>>(Wzi_r, Wzi_i, NZg, 0, 1.0f/NZg);
  build_dft<<<(64*64 + 255)/256, 256, 0, stream>>>(Wyi_r, Wyi_i, NYg, 0, 1.0f/NYg);
  build_dft<<<(64*64 + 255)/256, 256, 0, stream>>>(Wxi_r, Wxi_i, NXg, 0, 1.0f/NXg);
  build_dft<<<(32*32 + 255)/256, 256, 0, stream>>>(Wzf_r, Wzf_i, NZg, 1, 1.0f);
  build_dft<<<(64*64 + 255)/256, 256, 0, stream>>>(Wyf_r, Wyf_i, NYg, 1, 1.0f);
  build_dft<<<(64*64 + 255)/256, 256, 0, stream>>>(Wxf_r, Wxf_i, NXg, 1, 1.0f);

  // ---- inverse transform of C: z (W on B side, data contiguous in k), y, x ----
  launch_cgemm<1,2,0,1>(Ck_re, Ck_im, Wzi_r, Wzi_i, Are, Aim,
                        NMOM*NYg*NXg, NZg, NZg,
                        0, NZg, 1,   0, 1, NZg,   0, NZg, 1,
                        1024, 1, stream);
  launch_cgemm<4,1,1,0>(Wyi_r, Wyi_i, Are, Aim, Bre, Bim,
                        NYg, NXg*NZg, NYg,
                        0, NYg, 1,   (long)G, (long)NXg*NZg, 1,   (long)G, (long)NXg*NZg, 1,
                        4096, NMOM, stream);
  launch_cgemm<4,1,1,0>(Wxi_r, Wxi_i, Bre, Bim, Are, Aim,
                        NXg, NZg, NXg,
                        0, NXg, 1,   (long)NXg*NZg, NZg, 1,   (long)NXg*NZg, NZg, 1,
                        4096, NMOM*NYg, stream);

  // ---- inverse transform of F (6 EM fields) ----
  launch_cgemm<1,2,0,1>(Fk_re, Fk_im, Wzi_r, Wzi_i, Far, Fai,
                        NFLD*NYg*NXg, NZg, NZg,
                        0, NZg, 1,   0, 1, NZg,   0, NZg, 1,
                        1024, 1, stream);
  launch_cgemm<4,1,1,0>(Wyi_r, Wyi_i, Far, Fai, Fbr, Fbi,
                        NYg, NXg*NZg, NYg,
                        0, NYg, 1,   (long)G, (long)NXg*NZg, 1,   (long)G, (long)NXg*NZg, 1,
                        4096, NFLD, stream);
  launch_cgemm<4,1,1,0>(Wxi_r, Wxi_i, Fbr, Fbi, Far, Fai,
                        NXg, NZg, NXg,
                        0, NXg, 1,   (long)NXg*NZg, NZg, 1,   (long)NXg*NZg, NZg, 1,
                        4096, NFLD*NYg, stream);

  // ---- nonlinear term in real space: reads C(A), EB(Fa) -> writes B ----
  {
    long total = (long)NMOM * G;
    int blocks = (int)((total + 255) / 256);
    nl_kernel<<<blocks, 256, 0, stream>>>(Are, Aim, Far, Fai, Bre, Bim,
                                          qs, al, us, snp, snm, smp, smm, spp, spm);
  }

  // ---- forward transform of NL: z, y, x (ends in A) ----
  launch_cgemm<1,2,0,1>(Bre, Bim, Wzf_r, Wzf_i, Are, Aim,
                        NMOM*NYg*NXg, NZg, NZg,
                        0, NZg, 1,   0, 1, NZg,   0, NZg, 1,
                        1024, 1, stream);
  launch_cgemm<4,1,1,0>(Wyf_r, Wyf_i, Are, Aim, Bre, Bim,
                        NYg, NXg*NZg, NYg,
                        0, NYg, 1,   (long)G, (long)NXg*NZg, 1,   (long)G, (long)NXg*NZg, 1,
                        4096, NMOM, stream);
  launch_cgemm<4,1,1,0>(Wxf_r, Wxf_i, Bre, Bim, Are, Aim,
                        NXg, NZg, NXg,
                        0, NXg, 1,   (long)NXg*NZg, NZg, 1,   (long)NXg*NZg, NZg, 1,
                        4096, NMOM*NYg, stream);

  // ---- assemble dCk ----
  {
    long total = (long)NMOM * G;
    int blocks = (int)((total + 255) / 256);
    dck_kernel<<<blocks, 256, 0, stream>>>(Ck_re, Ck_im, Are, Aim,
                                           kxg, kyg, kzg, k2g, col, mask,
                                           nu, Dd, al, us,
                                           snp, snm, smp, smm, spp, spm, out);
  }

  // ---- assemble dFk ----
  {
    int blocks = (G + 255) / 256;
    dfk_kernel<<<blocks, 256, 0, stream>>>(Fk_re, Fk_im, Ck_re, Ck_im,
                                           kxg, kyg, kzg, qs, al, us, Om, out);
  }
}